// LuongAttnDecoderRNN_79199196938638
// MI455X (gfx1250) — compile-verified
//
#include <hip/hip_runtime.h>
#include <hip/hip_bf16.h>
#include <math.h>

#define Hh   512
#define Kk   128
#define Vv   32000
#define ANum 20
#define Nn   64
#define Bb   32
#define Tt   256
#define Ss   32
#define Aa   5
#define CW   (3*Hh + Kk)     // 1664 concat width
#define GW   (2*Hh + ANum)   // 1044 gate concat width
#define KC   32              // K-chunk staged in LDS per iteration

typedef __attribute__((ext_vector_type(2))) float v2f;
typedef __attribute__((ext_vector_type(8))) float v8f;
typedef __attribute__((ext_vector_type(4))) int   v4i;

#define AS1 __attribute__((address_space(1)))
#define AS3 __attribute__((address_space(3)))

#if defined(__has_builtin)
#  if __has_builtin(__builtin_amdgcn_global_load_async_to_lds_b128)
#    define HAVE_ASYNC_LDS 1
#  endif
#  if __has_builtin(__builtin_amdgcn_s_wait_asynccnt)
#    define HAVE_WAIT_ASYNC 1
#  endif
#endif
#ifndef HAVE_ASYNC_LDS
#  define HAVE_ASYNC_LDS 0
#endif
#ifndef HAVE_WAIT_ASYNC
#  define HAVE_WAIT_ASYNC 0
#endif

__device__ __forceinline__ void wait_async_zero() {
#if HAVE_WAIT_ASYNC
  __builtin_amdgcn_s_wait_asynccnt(0);
#else
  asm volatile("s_wait_asynccnt 0x0" ::: "memory");
#endif
}

// Pointer casts for the async-to-LDS DMA builtin:
//   global source:  int4 pointer in device (as1) space
//   LDS dest:       int4 pointer in shared (as3) space; flat LDS addr[31:0]
//                   is the wave-relative LDS offset per the CDNA5 aperture map.
__device__ __forceinline__ AS1 v4i* as_gbl16(const float* p) {
  return (AS1 v4i*)(uintptr_t)p;
}
__device__ __forceinline__ AS3 v4i* as_lds16(float* p) {
  return (AS3 v4i*)(unsigned)(uintptr_t)p;
}

__device__ __forceinline__ void stage_chunk(const float* g, float* l) {
#if HAVE_ASYNC_LDS
  __builtin_amdgcn_global_load_async_to_lds_b128(as_gbl16(g), as_lds16(l), 0, 0);
#else
  float4 tmp = *(const float4*)g;
  *(float4*)l = tmp;
#endif
}

// ---------------------------------------------------------------------------
// WMMA f32 GEMM:  C[M,N] = act( A[M,K] @ W[N,K]^T + bias[N] )
//   block = 256 threads = 8 waves arranged 2(M) x 4(N); block tile 32x256.
//   wave tile = 16(M) x 64(N): 4 accumulators of V_WMMA_F32_16X16X4_F32.
//   A chunks (32xKC) are double-buffered in LDS via
//   GLOBAL_LOAD_ASYNC_TO_LDS_B128 (one b128 per thread): the DMA for chunk
//   i+1 is issued right after the barrier of chunk i, overlapping the 32
//   WMMAs of chunk i; s_wait_asynccnt 0 at the top of the next iteration
//   retires it. One barrier per chunk.
//   Geometry: M % 32 == 0, N % 256 == 0, K % KC == 0 for every launch, so
//   there is no divergence and EXEC is all-1s at every WMMA (ISA requirement).
// Frag indexing (ISA 7.12.2): A/B frag k = k0 + 2*(lane>>4) + r;
//   C/D frag: vgpr r -> (m = r + 8*(lane>>4), n = lane&15).
// ---------------------------------------------------------------------------
template <int ACT> // 0 = none, 1 = tanh
__global__ void gemm_wmma_f32(const float* __restrict__ A,
                              const float* __restrict__ W,
                              const float* __restrict__ bias,
                              float* __restrict__ C,
                              int M, int N, int Kd, int ldc) {
  __shared__ float As[2][32 * KC]; // 2 x 4 KB A staging buffers

  const int wave  = threadIdx.x >> 5;
  const int lane  = threadIdx.x & 31;
  const int waveM = wave >> 2;   // 0..1
  const int waveN = wave & 3;    // 0..3
  const int half  = lane >> 4;
  const int nl    = lane & 15;

  const int mblk  = blockIdx.y * 32;
  const int ntile = blockIdx.x * 256 + waveN * 64;

  // B = W^T with W row-major (N x K): per-lane row pointers for 4 N-subtiles.
  const float* __restrict__ wrow0 = W + (size_t)(ntile +  0 + nl) * Kd;
  const float* __restrict__ wrow1 = W + (size_t)(ntile + 16 + nl) * Kd;
  const float* __restrict__ wrow2 = W + (size_t)(ntile + 32 + nl) * Kd;
  const float* __restrict__ wrow3 = W + (size_t)(ntile + 48 + nl) * Kd;

  // A staging map: thread -> one 16B segment of the 32xKC chunk.
  const int srow = threadIdx.x >> 3;  // 0..31
  const int sseg = threadIdx.x & 7;   // 0..7
  const float* __restrict__ gA = A + (size_t)(mblk + srow) * Kd + sseg * 4;
  float* lA0 = &As[0][srow * KC + sseg * 4];
  float* lA1 = &As[1][srow * KC + sseg * 4];

  v8f c0 = {}, c1 = {}, c2 = {}, c3 = {};

  const int nchunks = Kd / KC;
  stage_chunk(gA, lA0);                       // prologue: chunk 0 -> buf 0

  for (int i = 0; i < nchunks; ++i) {
    const int kc = i * KC;
#if HAVE_ASYNC_LDS
    wait_async_zero();                        // chunk i DMA retired (per-wave)
#endif
    __syncthreads();                          // chunk i visible to all waves;
                                              // all waves done reading i-1
    if (i + 1 < nchunks) {                    // prefetch chunk i+1 (overlaps)
      stage_chunk(gA + kc + KC, ((i + 1) & 1) ? lA1 : lA0);
    }

    const float* aBase = &As[i & 1][(waveM * 16 + nl) * KC];
#pragma unroll
    for (int ks = 0; ks < KC; ks += 4) {
      const int k  = ks + 2 * half;
      const int gk = kc + k;
      v2f a, b0, b1, b2, b3;
      a.x  = aBase[k];   a.y  = aBase[k + 1];
      b0.x = wrow0[gk];  b0.y = wrow0[gk + 1];
      b1.x = wrow1[gk];  b1.y = wrow1[gk + 1];
      b2.x = wrow2[gk];  b2.y = wrow2[gk + 1];
      b3.x = wrow3[gk];  b3.y = wrow3[gk + 1];
      c0 = __builtin_amdgcn_wmma_f32_16x16x4_f32(false, a, false, b0, (short)0, c0, false, false);
      c1 = __builtin_amdgcn_wmma_f32_16x16x4_f32(false, a, false, b1, (short)0, c1, false, false);
      c2 = __builtin_amdgcn_wmma_f32_16x16x4_f32(false, a, false, b2, (short)0, c2, false, false);
      c3 = __builtin_amdgcn_wmma_f32_16x16x4_f32(false, a, false, b3, (short)0, c3, false, false);
    }
  }

  const float bn0 = bias ? bias[ntile +  0 + nl] : 0.0f;
  const float bn1 = bias ? bias[ntile + 16 + nl] : 0.0f;
  const float bn2 = bias ? bias[ntile + 32 + nl] : 0.0f;
  const float bn3 = bias ? bias[ntile + 48 + nl] : 0.0f;
  const int mbase = mblk + waveM * 16 + 8 * half;
#pragma unroll
  for (int r = 0; r < 8; ++r) {
    const size_t rowoff = (size_t)(mbase + r) * ldc;
    float v0 = c0[r] + bn0, v1 = c1[r] + bn1, v2 = c2[r] + bn2, v3 = c3[r] + bn3;
    if (ACT == 1) { v0 = tanhf(v0); v1 = tanhf(v1); v2 = tanhf(v2); v3 = tanhf(v3); }
    C[rowoff + ntile +  0 + nl] = v0;
    C[rowoff + ntile + 16 + nl] = v1;
    C[rowoff + ntile + 32 + nl] = v2;
    C[rowoff + ntile + 48 + nl] = v3;
  }
}

// ---------------------------------------------------------------------------
// embedded[row, :] = emb[input_seq[row], :]   (row = t*B + b)
// ---------------------------------------------------------------------------
__global__ void k_embed(const int* __restrict__ seq, const float* __restrict__ emb,
                        float* __restrict__ embedded) {
  int idx = blockIdx.x * blockDim.x + threadIdx.x;
  if (idx >= Nn * Bb * Hh) return;
  int row = idx / Hh, j = idx % Hh;
  embedded[idx] = emb[(size_t)seq[row] * Hh + j];
}

// ---------------------------------------------------------------------------
// GRU pointwise gates for timestep t.
// ---------------------------------------------------------------------------
__global__ void k_gru_gates(const float* __restrict__ xg, const float* __restrict__ gh,
                            const float* __restrict__ bhh,
                            float* __restrict__ h, float* __restrict__ rnn,
                            float* __restrict__ concatbuf, float* __restrict__ hid_out,
                            int t) {
  int idx = blockIdx.x * blockDim.x + threadIdx.x;
  if (idx >= Bb * Hh) return;
  int b = idx / Hh, j = idx % Hh;
  const float* xgr = xg + (size_t)(t * Bb + b) * (3 * Hh);
  const float* ghr = gh + (size_t)b * (3 * Hh);
  float ir = xgr[j], iz = xgr[Hh + j], inn = xgr[2 * Hh + j];
  float hr = ghr[j] + bhh[j];
  float hz = ghr[Hh + j] + bhh[Hh + j];
  float hn = ghr[2 * Hh + j] + bhh[2 * Hh + j];
  float r = 1.0f / (1.0f + expf(-(ir + hr)));
  float z = 1.0f / (1.0f + expf(-(iz + hz)));
  float n = tanhf(inn + r * hn);
  float hnew = (1.0f - z) * n + z * h[idx];
  h[idx] = hnew;
  rnn[(size_t)(t * Bb + b) * Hh + j] = hnew;
  concatbuf[(size_t)(t * Bb + b) * CW + j] = hnew;   // concat col block 0
  if (t == Nn - 1) hid_out[idx] = hnew;              // hT output
}

// ---------------------------------------------------------------------------
// Rank-1 attention precompute: u1=Wh^T v, u2=We^T v, ua, ue, bv, bva.
// ---------------------------------------------------------------------------
__global__ void k_attn_prep(const float* __restrict__ attn_W, const float* __restrict__ attn_b,
                            const float* __restrict__ attn_v,
                            const float* __restrict__ attr_W, const float* __restrict__ attr_b,
                            const float* __restrict__ attr_v,
                            float* __restrict__ u1, float* __restrict__ u2,
                            float* __restrict__ ua, float* __restrict__ ue,
                            float* __restrict__ scal) {
  int idx = blockIdx.x * blockDim.x + threadIdx.x;
  if (idx < Hh) {
    float s = 0.f;
    for (int j = 0; j < Hh; ++j) s += attn_W[(size_t)j * (2 * Hh) + idx] * attn_v[j];
    u1[idx] = s;
  } else if (idx < 2 * Hh) {
    int k = idx - Hh; float s = 0.f;
    for (int j = 0; j < Hh; ++j) s += attn_W[(size_t)j * (2 * Hh) + Hh + k] * attn_v[j];
    u2[k] = s;
  } else if (idx < 3 * Hh) {
    int k = idx - 2 * Hh; float s = 0.f;
    for (int j = 0; j < Hh; ++j) s += attr_W[(size_t)j * (Hh + Kk) + k] * attr_v[j];
    ua[k] = s;
  } else if (idx < 3 * Hh + Kk) {
    int k = idx - 3 * Hh; float s = 0.f;
    for (int j = 0; j < Hh; ++j) s += attr_W[(size_t)j * (Hh + Kk) + Hh + k] * attr_v[j];
    ue[k] = s;
  } else if (idx == 3 * Hh + Kk) {
    float s = 0.f; for (int j = 0; j < Hh; ++j) s += attn_b[j] * attn_v[j]; scal[0] = s;
  } else if (idx == 3 * Hh + Kk + 1) {
    float s = 0.f; for (int j = 0; j < Hh; ++j) s += attr_b[j] * attr_v[j]; scal[1] = s;
  }
}

// out[i] = dot(X[i, 0:len], u) + (scalIdx >= 0 ? scal[scalIdx] : 0)
__global__ void k_rowdot(const float* __restrict__ X, const float* __restrict__ u,
                         const float* __restrict__ scal, int scalIdx,
                         int rows, int len, float* __restrict__ out) {
  int i = blockIdx.x * blockDim.x + threadIdx.x;
  if (i >= rows) return;
  const float* x = X + (size_t)i * len;
  float s = (scalIdx >= 0) ? scal[scalIdx] : 0.0f;
  for (int k = 0; k < len; ++k) s += x[k] * u[k];
  out[i] = s;
}

// aw[b,n,t] = softmax_t( tanh(a[n,b] + c[t,b]) );  one block per (b,n)
__global__ void k_attn_softmax(const float* __restrict__ a, const float* __restrict__ c,
                               float* __restrict__ aw, int L) {
  __shared__ float buf[256];
  int b = blockIdx.x / Nn, n = blockIdx.x % Nn;
  int t = threadIdx.x;
  float av = a[n * Bb + b];
  float e = (t < L) ? tanhf(av + c[t * Bb + b]) : -INFINITY;
  buf[t] = e; __syncthreads();
  for (int s = blockDim.x >> 1; s > 0; s >>= 1) {
    if (t < s) buf[t] = fmaxf(buf[t], buf[t + s]);
    __syncthreads();
  }
  float mx = buf[0]; __syncthreads();
  float ex = (t < L) ? expf(e - mx) : 0.0f;
  buf[t] = ex; __syncthreads();
  for (int s = blockDim.x >> 1; s > 0; s >>= 1) {
    if (t < s) buf[t] += buf[t + s];
    __syncthreads();
  }
  float sum = buf[0];
  if (t < L) aw[((size_t)b * Nn + n) * L + t] = ex / sum;
}

// dst[row, colOff + h] = sum_t aw[b,n,t] * enc[t,b,h]   (row = n*B + b)
__global__ void k_ctx(const float* __restrict__ aw, const float* __restrict__ enc,
                      float* __restrict__ dst, int L, int D, int dstStride, int colOff) {
  int idx = blockIdx.x * blockDim.x + threadIdx.x;
  if (idx >= Nn * Bb * D) return;
  int row = idx / D, hc = idx % D;
  int n = row / Bb, b = row % Bb;
  const float* awp = aw + ((size_t)b * Nn + n) * L;
  float s = 0.f;
  for (int t = 0; t < L; ++t) s += awp[t] * enc[((size_t)t * Bb + b) * D + hc];
  dst[(size_t)row * dstStride + colOff + hc] = s;
}

// e4p[b,j] = proj_b[j] + cat(enc4[0,b,:], enc4[1,b,:]) . proj_W[j,:]
__global__ void k_e4proj(const float* __restrict__ enc4, const float* __restrict__ pW,
                         const float* __restrict__ pb, float* __restrict__ e4p) {
  int idx = blockIdx.x * blockDim.x + threadIdx.x;
  if (idx >= Bb * ANum) return;
  int b = idx / ANum, j = idx % ANum;
  float s = pb[j];
  for (int k = 0; k < ANum; ++k) s += enc4[(size_t)b * ANum + k] * pW[j * 2 * ANum + k];
  for (int k = 0; k < ANum; ++k) s += enc4[(size_t)(Bb + b) * ANum + k] * pW[j * 2 * ANum + ANum + k];
  e4p[idx] = s;
}

// gate[row,j] = tanh( cat(embedded,rnn,e4p) . gate_W[j,:] + gate_b[j] )
__global__ void k_gate(const float* __restrict__ embedded, const float* __restrict__ rnn,
                       const float* __restrict__ e4p, const float* __restrict__ gWt,
                       const float* __restrict__ gb, float* __restrict__ gate_out) {
  int idx = blockIdx.x * blockDim.x + threadIdx.x;
  if (idx >= Nn * Bb * ANum) return;
  int row = idx / ANum, j = idx % ANum;
  int b = row % Bb;
  const float* w  = gWt + (size_t)j * GW;
  const float* er = embedded + (size_t)row * Hh;
  const float* rr = rnn + (size_t)row * Hh;
  float s = gb[j];
  for (int k = 0; k < Hh; ++k) s += er[k] * w[k];
  for (int k = 0; k < Hh; ++k) s += rr[k] * w[Hh + k];
  for (int k = 0; k < ANum; ++k) s += e4p[b * ANum + k] * w[2 * Hh + k];
  gate_out[idx] = tanhf(s);
}

// output[row, aspect_ids[i]] += gate[row, i % 20]  (duplicate ids -> atomics)
__global__ void k_scatter(const float* __restrict__ gate_out, const int* __restrict__ ids,
                          float* __restrict__ out) {
  long idx = (long)blockIdx.x * blockDim.x + threadIdx.x;
  const long total = (long)Nn * Bb * 100 * ANum;
  if (idx >= total) return;
  int i = (int)(idx % (100 * ANum));
  int row = (int)(idx / (100 * ANum));
  float val = gate_out[row * ANum + (i % ANum)];
  atomicAdd(&out[(size_t)row * Vv + ids[i]], val);
}

// ---------------------------------------------------------------------------
extern "C" void kernel_launch(void* const* d_in, const int* in_sizes, int n_in,
                              void* d_out, int out_size, void* d_ws, size_t ws_size,
                              hipStream_t stream) {
  const int*   input_seq   = (const int*)  d_in[0];
  const float* last_hidden = (const float*)d_in[1];
  const float* enc1        = (const float*)d_in[2];
  const float* enc2        = (const float*)d_in[3];
  const float* enc3        = (const float*)d_in[4];
  const float* enc4        = (const float*)d_in[5];
  const int*   aspect_ids  = (const int*)  d_in[6];
  const float* emb         = (const float*)d_in[7];
  const float* gru_Wih     = (const float*)d_in[8];
  const float* gru_Whh     = (const float*)d_in[9];
  const float* gru_bih     = (const float*)d_in[10];
  const float* gru_bhh     = (const float*)d_in[11];
  const float* attn_W      = (const float*)d_in[12];
  const float* attn_b      = (const float*)d_in[13];
  const float* attn_v      = (const float*)d_in[14];
  const float* attr_W      = (const float*)d_in[15];
  const float* attr_b      = (const float*)d_in[16];
  const float* attr_v      = (const float*)d_in[17];
  const float* concat_W    = (const float*)d_in[18];
  const float* concat_b    = (const float*)d_in[19];
  const float* out_W       = (const float*)d_in[20];
  const float* out_b       = (const float*)d_in[21];
  const float* gate_W      = (const float*)d_in[22];
  const float* gate_b      = (const float*)d_in[23];
  const float* proj_W      = (const float*)d_in[24];
  const float* proj_b      = (const float*)d_in[25];

  // ---- workspace layout (floats) ----
  float* ws = (float*)d_ws;
  float* f_embedded = ws;                          // 2048*512
  float* f_xg       = f_embedded + 1048576;        // 2048*1536
  float* f_rnn      = f_xg       + 3145728;        // 2048*512
  float* f_h        = f_rnn      + 1048576;        // 32*512
  float* f_gh       = f_h        + 16384;          // 32*1536
  float* f_u1       = f_gh       + 49152;          // 512
  float* f_u2       = f_u1       + 512;            // 512
  float* f_ua       = f_u2       + 512;            // 512
  float* f_ue       = f_ua       + 512;            // 128
  float* f_scal     = f_ue       + 128;            // 2 (padded)
  float* f_a1       = f_scal     + 128;            // 2048
  float* f_a3       = f_a1       + 2048;           // 2048
  float* f_c1       = f_a3       + 2048;           // 8192
  float* f_c2       = f_c1       + 8192;           // 1024
  float* f_c3       = f_c2       + 1024;           // 256 (160 used)
  float* f_concat   = f_c3       + 256;            // 2048*1664
  float* f_cout     = f_concat   + 3407872;        // 2048*512
  float* f_e4p      = f_cout     + 1048576;        // 640

  // ---- output layout (floats, concatenated return tuple) ----
  float* out        = (float*)d_out;
  float* out_output = out;                                   // (N,B,V)
  float* out_hidden = out_output + (size_t)Nn * Bb * Vv;     // (1,B,H)
  float* out_aw1    = out_hidden + (size_t)Bb * Hh;          // (B,N,T)
  float* out_aw2    = out_aw1    + (size_t)Bb * Nn * Tt;     // (B,N,S)
  float* out_aw3    = out_aw2    + (size_t)Bb * Nn * Ss;     // (B,N,A)
  float* out_gate   = out_aw3    + (size_t)Bb * Nn * Aa;     // (N,B,ANUM)

  // 1) embedding gather
  k_embed<<<(Nn * Bb * Hh + 255) / 256, 256, 0, stream>>>(input_seq, emb, f_embedded);

  // 2) all-timestep input gates: xg = embedded @ Wih^T + bih  (2048x512 @ 512x1536)
  gemm_wmma_f32<0><<<dim3(6, 64), 256, 0, stream>>>(
      f_embedded, gru_Wih, gru_bih, f_xg, Nn * Bb, 3 * Hh, Hh, 3 * Hh);

  // 3) h0 = last_hidden
  (void)hipMemcpyAsync(f_h, last_hidden, (size_t)Bb * Hh * sizeof(float),
                       hipMemcpyDeviceToDevice, stream);

  // 4) sequential GRU: per step, gh = h @ Whh^T (WMMA), then pointwise gates
  for (int t = 0; t < Nn; ++t) {
    gemm_wmma_f32<0><<<dim3(6, 1), 256, 0, stream>>>(
        f_h, gru_Whh, nullptr, f_gh, Bb, 3 * Hh, Hh, 3 * Hh);
    k_gru_gates<<<(Bb * Hh + 255) / 256, 256, 0, stream>>>(
        f_xg, f_gh, gru_bhh, f_h, f_rnn, f_concat, out_hidden, t);
  }

  // 5) rank-1 attention precompute
  k_attn_prep<<<(3 * Hh + Kk + 2 + 255) / 256, 256, 0, stream>>>(
      attn_W, attn_b, attn_v, attr_W, attr_b, attr_v, f_u1, f_u2, f_ua, f_ue, f_scal);

  // 6) scores: a = rnn.u + b.v ; c = enc.u2
  k_rowdot<<<(Nn * Bb + 255) / 256, 256, 0, stream>>>(f_rnn, f_u1, f_scal, 0, Nn * Bb, Hh, f_a1);
  k_rowdot<<<(Nn * Bb + 255) / 256, 256, 0, stream>>>(f_rnn, f_ua, f_scal, 1, Nn * Bb, Hh, f_a3);
  k_rowdot<<<(Tt * Bb + 255) / 256, 256, 0, stream>>>(enc1, f_u2, f_scal, -1, Tt * Bb, Hh, f_c1);
  k_rowdot<<<(Ss * Bb + 255) / 256, 256, 0, stream>>>(enc2, f_u2, f_scal, -1, Ss * Bb, Hh, f_c2);
  k_rowdot<<<(Aa * Bb + 255) / 256, 256, 0, stream>>>(enc3, f_ue, f_scal, -1, Aa * Bb, Kk, f_c3);

  // 7) softmax -> aw outputs, then context vectors written into concat buffer
  k_attn_softmax<<<Bb * Nn, 256, 0, stream>>>(f_a1, f_c1, out_aw1, Tt);
  k_attn_softmax<<<Bb * Nn, 32, 0, stream>>>(f_a1, f_c2, out_aw2, Ss);
  k_attn_softmax<<<Bb * Nn, 32, 0, stream>>>(f_a3, f_c3, out_aw3, Aa);
  k_ctx<<<(Nn * Bb * Hh + 255) / 256, 256, 0, stream>>>(out_aw1, enc1, f_concat, Tt, Hh, CW, Hh);
  k_ctx<<<(Nn * Bb * Hh + 255) / 256, 256, 0, stream>>>(out_aw2, enc2, f_concat, Ss, Hh, CW, 2 * Hh);
  k_ctx<<<(Nn * Bb * Kk + 255) / 256, 256, 0, stream>>>(out_aw3, enc3, f_concat, Aa, Kk, CW, 3 * Hh);

  // 8) concat_output = tanh(concat @ concat_W^T + b)  (2048x1664 @ 1664x512, WMMA)
  gemm_wmma_f32<1><<<dim3(2, 64), 256, 0, stream>>>(
      f_concat, concat_W, concat_b, f_cout, Nn * Bb, Hh, CW, Hh);

  // 9) vocab projection: output = concat_out @ out_W^T + out_b  (2048x512 @ 512x32000, WMMA)
  gemm_wmma_f32<0><<<dim3(125, 64), 256, 0, stream>>>(
      f_cout, out_W, out_b, out_output, Nn * Bb, Vv, Hh, Vv);

  // 10) gate path
  k_e4proj<<<(Bb * ANum + 63) / 64, 64, 0, stream>>>(enc4, proj_W, proj_b, f_e4p);
  k_gate<<<(Nn * Bb * ANum + 255) / 256, 256, 0, stream>>>(
      f_embedded, f_rnn, f_e4p, gate_W, gate_b, out_gate);

  // 11) scatter-add tiled gate values into vocab logits
  long total = (long)Nn * Bb * 100 * ANum;
  k_scatter<<<(int)((total + 255) / 256), 256, 0, stream>>>(out_gate, aspect_ids, out_output);
}